// EMEstimator_36696200577471
// MI455X (gfx1250) — compile-verified
//
#include <hip/hip_runtime.h>

#define NSRC 4
#define MDIM 4
#define TFR  512
#define NIT  3
#define EPSV 1e-6f
#define NSYM 10
#define BDS  18   // block-diagonal 16x16 LDS row stride (72B: pairs stay 8B-aligned)
#define FST  36   // staging row stride: 32 frames + pad (144B, pairs 8B-aligned)

typedef float v2f __attribute__((ext_vector_type(2)));
typedef float v8f __attribute__((ext_vector_type(8)));

__device__ __forceinline__ int symIdx(int i, int j) {  // i<=j
    return i * (7 - i) / 2 + j;
}

__global__ __launch_bounds__(256) void em_kernel(const float* __restrict__ lm,
                                                 const float* __restrict__ x,
                                                 float* __restrict__ out) {
    const int bf   = blockIdx.x;
    const int tid  = threadIdx.x;
    const int lane = tid & 31;
    const int wav  = tid >> 5;
    const int rowi = lane & 15;   // WMMA A-row M / B,D-column N
    const int kh   = lane >> 4;   // K half: VGPR pair holds K={2kh, 2kh+1}

    __shared__ __align__(16) float Hl[64];           // H: [src][4][4]
    __shared__ __align__(16) float Gl[40];           // G accum: [src][10] symmetric
    __shared__ __align__(16) float HB[16 * BDS];     // block-diag H (symmetric)
    __shared__ __align__(16) float GB[16 * BDS];     // block-diag G/T (symmetric)
    __shared__ __align__(16) float Pb[16 * BDS];     // P = HB*GB
    __shared__ __align__(16) float Qb[16 * BDS];     // Q = P*HB
    __shared__ __align__(16) float Lst[8][16][FST];  // per-wave lm stage [src][frame]
    __shared__ __align__(16) float Est[8][16][FST];  // per-wave A-entry stage [entry][frame]

    const float* lmp = lm + (size_t)bf * NSRC * TFR;
    const float* xp  = x  + (size_t)bf * MDIM * TFR;

    // H init = identity per source
    if (tid < 64) {
        int rc = tid & 15;
        Hl[tid] = ((rc >> 2) == (rc & 3)) ? 1.0f : 0.0f;
    }

    for (int it = 0; it < NIT; ++it) {
        if (tid < 40) Gl[tid] = 0.0f;
        __syncthreads();

        // symmetric H entries, uniform across block
        float Hs[NSRC][NSYM];
        #pragma unroll
        for (int k = 0; k < NSRC; ++k) {
            int e = 0;
            #pragma unroll
            for (int i = 0; i < 4; ++i)
                #pragma unroll
                for (int j = i; j < 4; ++j)
                    Hs[k][e++] = Hl[k * 16 + i * 4 + j];
        }

        // per-wave WMMA accumulator: C[src][entry] partial of G
        v8f creg = {0.f, 0.f, 0.f, 0.f, 0.f, 0.f, 0.f, 0.f};

        #pragma unroll
        for (int fr = 0; fr < TFR / 256; ++fr) {
            const int t = tid + fr * 256;
            float lmv[NSRC], xv[MDIM];
            #pragma unroll
            for (int k = 0; k < NSRC; ++k) lmv[k] = lmp[k * TFR + t];
            #pragma unroll
            for (int m = 0; m < MDIM; ++m) xv[m] = xp[m * TFR + t];

            // Y = sum_k lm_k H_k + eps*I (10 symmetric entries)
            float ys[NSYM];
            #pragma unroll
            for (int e = 0; e < NSYM; ++e) {
                float v = lmv[0] * Hs[0][e];
                v = fmaf(lmv[1], Hs[1][e], v);
                v = fmaf(lmv[2], Hs[2][e], v);
                v = fmaf(lmv[3], Hs[3][e], v);
                ys[e] = v;
            }
            ys[0] += EPSV; ys[4] += EPSV; ys[7] += EPSV; ys[9] += EPSV;

            float mm[16];
            mm[0]=ys[0];  mm[1]=ys[1];  mm[2]=ys[2];  mm[3]=ys[3];
            mm[4]=ys[1];  mm[5]=ys[4];  mm[6]=ys[5];  mm[7]=ys[6];
            mm[8]=ys[2];  mm[9]=ys[5];  mm[10]=ys[7]; mm[11]=ys[8];
            mm[12]=ys[3]; mm[13]=ys[6]; mm[14]=ys[8]; mm[15]=ys[9];

            // 4x4 inverse via cofactors
            float s0 = mm[0]*mm[5] - mm[4]*mm[1];
            float s1 = mm[0]*mm[6] - mm[4]*mm[2];
            float s2 = mm[0]*mm[7] - mm[4]*mm[3];
            float s3 = mm[1]*mm[6] - mm[5]*mm[2];
            float s4 = mm[1]*mm[7] - mm[5]*mm[3];
            float s5 = mm[2]*mm[7] - mm[6]*mm[3];
            float c5 = mm[10]*mm[15] - mm[14]*mm[11];
            float c4 = mm[9]*mm[15]  - mm[13]*mm[11];
            float c3 = mm[9]*mm[14]  - mm[13]*mm[10];
            float c2 = mm[8]*mm[15]  - mm[12]*mm[11];
            float c1 = mm[8]*mm[14]  - mm[12]*mm[10];
            float c0 = mm[8]*mm[13]  - mm[12]*mm[9];
            float det = s0*c5 - s1*c4 + s2*c3 + s3*c2 - s4*c1 + s5*c0;
            float id  = 1.0f / det;

            float inv[16];
            inv[0]  = ( mm[5]*c5 - mm[6]*c4 + mm[7]*c3) * id;
            inv[1]  = (-mm[1]*c5 + mm[2]*c4 - mm[3]*c3) * id;
            inv[2]  = ( mm[13]*s5 - mm[14]*s4 + mm[15]*s3) * id;
            inv[3]  = (-mm[9]*s5 + mm[10]*s4 - mm[11]*s3) * id;
            inv[4]  = (-mm[4]*c5 + mm[6]*c2 - mm[7]*c1) * id;
            inv[5]  = ( mm[0]*c5 - mm[2]*c2 + mm[3]*c1) * id;
            inv[6]  = (-mm[12]*s5 + mm[14]*s2 - mm[15]*s1) * id;
            inv[7]  = ( mm[8]*s5 - mm[10]*s2 + mm[11]*s1) * id;
            inv[8]  = ( mm[4]*c4 - mm[5]*c2 + mm[7]*c0) * id;
            inv[9]  = (-mm[0]*c4 + mm[1]*c2 - mm[3]*c0) * id;
            inv[10] = ( mm[12]*s4 - mm[13]*s2 + mm[15]*s0) * id;
            inv[11] = (-mm[8]*s4 + mm[9]*s2 - mm[11]*s0) * id;
            inv[12] = (-mm[4]*c3 + mm[5]*c1 - mm[6]*c0) * id;
            inv[13] = ( mm[0]*c3 - mm[1]*c1 + mm[2]*c0) * id;
            inv[14] = (-mm[12]*s3 + mm[13]*s1 - mm[14]*s0) * id;
            inv[15] = ( mm[8]*s3 - mm[9]*s1 + mm[10]*s0) * id;

            // Yix = Yi * x
            float yix[4];
            #pragma unroll
            for (int i = 0; i < 4; ++i) {
                float v = inv[i*4+0] * xv[0];
                v = fmaf(inv[i*4+1], xv[1], v);
                v = fmaf(inv[i*4+2], xv[2], v);
                v = fmaf(inv[i*4+3], xv[3], v);
                yix[i] = v;
            }

            // stage this frame (lane) for the wave's WMMA reduction:
            // Lst[w][src][frame], Est[w][entry][frame] (transposed for b64 reads)
            #pragma unroll
            for (int k = 0; k < NSRC; ++k) Lst[wav][k][lane] = lmv[k];
            {
                int e = 0;
                #pragma unroll
                for (int i = 0; i < 4; ++i)
                    #pragma unroll
                    for (int j = i; j < 4; ++j) {
                        Est[wav][e][lane] = yix[i] * yix[j] - inv[i*4+j];
                        ++e;
                    }
            }
            asm volatile("s_wait_dscnt 0" ::: "memory");

            // G partial: C[src][entry] += lm(16x4-slice) x A-entries(4x16-slice)
            // over this wave's 32 frames = 8 chained V_WMMA_F32_16X16X4_F32
            #pragma unroll
            for (int q = 0; q < 8; ++q) {
                const int fc = 4 * q + 2 * kh;
                const v2f a_ = *(const v2f*)&Lst[wav][rowi][fc];
                const v2f b_ = *(const v2f*)&Est[wav][rowi][fc];
                creg = __builtin_amdgcn_wmma_f32_16x16x4_f32(
                    false, a_, false, b_, (short)0, creg, false, false);
            }
        }

        // combine wave partials: rows 0..3 (sources) x cols 0..9 (entries)
        if (lane < NSYM) {
            #pragma unroll
            for (int v = 0; v < NSRC; ++v)
                atomicAdd(&Gl[v * NSYM + lane], creg[v]);
        }
        __syncthreads();

        // Expand block-diagonal 16x16 operands into LDS (uniform loads + cndmask)
        {
            const int r = tid >> 4, c = tid & 15;
            const int diag = ((r >> 2) == (c >> 2));
            const float hv = Hl[(r >> 2) * 16 + (r & 3) * 4 + (c & 3)];
            const int i = r & 3, j = c & 3;
            const int lo = i < j ? i : j, hi = i < j ? j : i;
            const float gv = Gl[(r >> 2) * NSYM + symIdx(lo, hi)] * (1.0f / (float)TFR);
            HB[r * BDS + c] = diag ? hv : 0.0f;
            GB[r * BDS + c] = diag ? gv : 0.0f;
        }
        __syncthreads();

        // H <- H + HBD*GBD*HBD via two chains of four V_WMMA_F32_16X16X4_F32.
        // HBD/GBD symmetric => column reads become contiguous row b64 reads.
        if (tid < 32) {
            // P = HBD x GBD
            v8f acc = {0.f, 0.f, 0.f, 0.f, 0.f, 0.f, 0.f, 0.f};
            #pragma unroll
            for (int jj = 0; jj < 4; ++jj) {
                const int kb = 4 * jj + 2 * kh;
                const v2f a_ = *(const v2f*)&HB[rowi * BDS + kb];
                const v2f b_ = *(const v2f*)&GB[rowi * BDS + kb];
                acc = __builtin_amdgcn_wmma_f32_16x16x4_f32(
                    false, a_, false, b_, (short)0, acc, false, false);
            }
            #pragma unroll
            for (int v = 0; v < 8; ++v)
                Pb[(v + 8 * kh) * BDS + rowi] = acc[v];
            asm volatile("s_wait_dscnt 0" ::: "memory");

            // Q = P x HBD
            v8f acc2 = {0.f, 0.f, 0.f, 0.f, 0.f, 0.f, 0.f, 0.f};
            #pragma unroll
            for (int jj = 0; jj < 4; ++jj) {
                const int kb = 4 * jj + 2 * kh;
                const v2f a_ = *(const v2f*)&Pb[rowi * BDS + kb];
                const v2f b_ = *(const v2f*)&HB[rowi * BDS + kb];
                acc2 = __builtin_amdgcn_wmma_f32_16x16x4_f32(
                    false, a_, false, b_, (short)0, acc2, false, false);
            }
            #pragma unroll
            for (int v = 0; v < 8; ++v)
                Qb[(v + 8 * kh) * BDS + rowi] = acc2[v];
        }
        __syncthreads();

        // H += diagonal blocks of Q (uniform update by threads 0..63)
        if (tid < 64) {
            const int k = tid >> 4, i = (tid >> 2) & 3, j = tid & 3;
            Hl[tid] += Qb[(k * 4 + i) * BDS + (k * 4 + j)];
        }
        __syncthreads();
    }

    if (tid < 64) out[(size_t)bf * 64 + tid] = Hl[tid];
}

extern "C" void kernel_launch(void* const* d_in, const int* in_sizes, int n_in,
                              void* d_out, int out_size, void* d_ws, size_t ws_size,
                              hipStream_t stream) {
    const float* lm = (const float*)d_in[0];   // (B,F,N,T) f32
    const float* x  = (const float*)d_in[1];   // (B,F,M,T) f32
    float* out      = (float*)d_out;           // (B,F,N,M,M) f32

    const int BF = in_sizes[1] / (MDIM * TFR); // B*F independent problems

    em_kernel<<<dim3(BF), dim3(256), 0, stream>>>(lm, x, out);

    (void)n_in; (void)out_size; (void)d_ws; (void)ws_size;
}